// DeformConvAddReLU2d_15719580304168
// MI455X (gfx1250) — compile-verified
//
#include <hip/hip_runtime.h>

typedef __attribute__((ext_vector_type(2))) float v2f;
typedef __attribute__((ext_vector_type(8))) float v8f;

#define N_    8
#define CIN   128
#define H_    56
#define W_    56
#define COUT  128
#define KH    3
#define KW    3
#define HO_   56
#define WO_   56
#define HW    (H_ * W_)     // 3136
#define HWO   (HO_ * WO_)   // 3136
#define KTAPS (KH * KW)     // 9
#define TILE_M 16
#define ELD_STRIDE 20       // padded LDS stride for epilogue transpose

// ---------------------------------------------------------------------------
// Pre-pass: weight[cout][c][k] -> wP[k][c/2][cout][2]  (K-pair interleaved)
// so a WMMA B-fragment (c, c+1 for one cout) is ONE coalesced global_load_b64.
// ---------------------------------------------------------------------------
__global__ __launch_bounds__(256) void wtrans_kernel(const float* __restrict__ w,
                                                     float* __restrict__ wp) {
    int idx = blockIdx.x * 256 + threadIdx.x;   // ((k*64 + cp)*COUT + cout)*2 + s
    if (idx >= KTAPS * CIN * COUT) return;
    int s    = idx & 1;
    int t    = idx >> 1;
    int cout = t % COUT;
    int cp   = (t / COUT) % (CIN / 2);
    int k    = t / (COUT * (CIN / 2));
    int c    = cp * 2 + s;
    wp[idx]  = w[(cout * CIN + c) * KTAPS + k];
}

// ---------------------------------------------------------------------------
// Main kernel: one block (8 wave32) computes 16 positions x 128 couts.
// Implicit GEMM via V_WMMA_F32_16X16X4_F32 (full fp32, matches reference).
// ---------------------------------------------------------------------------
__global__ __launch_bounds__(256) void deform_conv_add_relu_kernel(
    const float* __restrict__ x,       // [N, CIN, H, W]
    const float* __restrict__ offset,  // [N, 2K, HO, WO]
    const float* __restrict__ wp,      // [K, CIN/2, COUT, 2]
    const float* __restrict__ bias,    // [COUT]
    const float* __restrict__ x2,      // [N, COUT, HO, WO]
    float* __restrict__ out)           // [N, COUT, HO, WO]
{
    // cols[buf][cpair][pos][2] : double-buffered, 2 x 8 KB
    __shared__ float cols[2][(CIN / 2) * TILE_M * 2];
    // epilogue transpose tile, padded stride
    __shared__ float eld[COUT * ELD_STRIDE];

    const int tid  = threadIdx.x;
    const int lane = tid & 31;
    const int wave = tid >> 5;             // 0..7 -> 16-cout slice

    // ---- gather-phase role: fixed position, 8 channels strided by 16 ----
    const int posg = tid & 15;
    const int cg0  = tid >> 4;             // 0..15

    const int m0  = blockIdx.x * TILE_M;   // tile start, flattened (n,ho,wo)
    const int n_t = m0 / HWO;              // block-uniform (tiles never straddle n)
    const int sp0 = m0 % HWO;              // block-uniform
    const int sp  = sp0 + posg;
    const int ho  = sp / WO_;
    const int wo  = sp % WO_;

    const float* __restrict__ xn = x + (size_t)n_t * CIN * HW;

    // ---- WMMA fragment indexing (ISA 7.12.2, wave32) ----
    const int am    = lane & 15;   // A: row M | B,D: col N
    const int khalf = lane >> 4;   // 0: K={0,1} | 1: K={2,3}
    const int cout0 = wave * 16;

    v8f acc = {0.f, 0.f, 0.f, 0.f, 0.f, 0.f, 0.f, 0.f};

    for (int k = 0; k < KTAPS; ++k) {
        float* __restrict__ colsb = cols[k & 1];

        // ================= gather phase =================
        const int ki = k / KW;
        const int kj = k % KW;
        const float offy = offset[(((size_t)n_t * (2 * KTAPS) + 2 * k) * HO_ + ho) * WO_ + wo];
        const float offx = offset[(((size_t)n_t * (2 * KTAPS) + 2 * k + 1) * HO_ + ho) * WO_ + wo];
        const float py = (float)(ho - 1 + ki) + offy;
        const float px = (float)(wo - 1 + kj) + offx;

        const float fy = floorf(py), fx = floorf(px);
        const int   y0 = (int)fy,    x0 = (int)fx;
        const float wy = py - fy,    wx = px - fx;
        const int   y1 = y0 + 1,     x1 = x0 + 1;

        const bool vy0 = (y0 >= 0) && (y0 < H_);
        const bool vy1 = (y1 >= 0) && (y1 < H_);
        const bool vx0 = (x0 >= 0) && (x0 < W_);
        const bool vx1 = (x1 >= 0) && (x1 < W_);

        const int cy0 = y0 < 0 ? 0 : (y0 > H_ - 1 ? H_ - 1 : y0);
        const int cy1 = y1 < 0 ? 0 : (y1 > H_ - 1 ? H_ - 1 : y1);
        const int cx0 = x0 < 0 ? 0 : (x0 > W_ - 1 ? W_ - 1 : x0);
        const int cx1 = x1 < 0 ? 0 : (x1 > W_ - 1 ? W_ - 1 : x1);

        const float w00 = (1.f - wy) * (1.f - wx) * ((vy0 && vx0) ? 1.f : 0.f);
        const float w01 = (1.f - wy) * wx         * ((vy0 && vx1) ? 1.f : 0.f);
        const float w10 = wy         * (1.f - wx) * ((vy1 && vx0) ? 1.f : 0.f);
        const float w11 = wy         * wx         * ((vy1 && vx1) ? 1.f : 0.f);

        const int i00 = cy0 * W_ + cx0;
        const int i01 = cy0 * W_ + cx1;
        const int i10 = cy1 * W_ + cx0;
        const int i11 = cy1 * W_ + cx1;

        #pragma unroll
        for (int j = 0; j < 8; ++j) {
            const int c = cg0 + j * 16;
            const float* __restrict__ xc = xn + (size_t)c * HW;
            const float v = w00 * xc[i00] + w01 * xc[i01]
                          + w10 * xc[i10] + w11 * xc[i11];
            // interleaved: [c/2][pos][c&1]; wave writes 32 consecutive dwords
            colsb[(c >> 1) * (TILE_M * 2) + posg * 2 + (c & 1)] = v;
        }
        __syncthreads();   // single barrier/tap: double buffer covers WAR on cols

        // ================= WMMA phase =================
        const v2f* __restrict__ wtk = (const v2f*)(wp + (size_t)k * (CIN / 2) * COUT * 2);
        #pragma unroll 8
        for (int kb = 0; kb < CIN; kb += 4) {
            const int cp = (kb >> 1) + khalf;        // K-pair index for this half-wave
            // A 16x4: one ds_load_b64 (channels cp*2, cp*2+1 at row am)
            v2f a = *(const v2f*)&colsb[cp * (TILE_M * 2) + am * 2];
            // B 4x16: one coalesced global_load_b64
            v2f b = wtk[(size_t)cp * COUT + cout0 + am];
            acc = __builtin_amdgcn_wmma_f32_16x16x4_f32(
                false, a, false, b, (short)0, acc, false, false);
        }
    }

    // ================= epilogue: +bias via LDS transpose, +x2, ReLU =================
    // D 16x16 f32: lane holds col N = am (cout), VGPR r holds row M = r + 8*khalf
    const float bv = bias[cout0 + am];
    __syncthreads();   // all waves done with cols[] reads; safe to repurpose LDS timeline
    #pragma unroll
    for (int r = 0; r < 8; ++r)
        eld[(cout0 + am) * ELD_STRIDE + 8 * khalf + r] = acc[r] + bv;
    __syncthreads();

    // coalesced read-modify-write: consecutive lanes -> consecutive spatial positions
    const int pos = tid & 15;
    const int cb  = tid >> 4;
    #pragma unroll
    for (int j = 0; j < 8; ++j) {
        const int cout = cb + 16 * j;
        const size_t gi = ((size_t)n_t * COUT + cout) * HWO + sp0 + pos;
        const float v = eld[cout * ELD_STRIDE + pos] + x2[gi];
        out[gi] = v > 0.f ? v : 0.f;
    }
}

// ---------------------------------------------------------------------------
extern "C" void kernel_launch(void* const* d_in, const int* in_sizes, int n_in,
                              void* d_out, int out_size, void* d_ws, size_t ws_size,
                              hipStream_t stream) {
    const float* x      = (const float*)d_in[0];   // [8,128,56,56]
    const float* offset = (const float*)d_in[1];   // [8,18,56,56]
    const float* weight = (const float*)d_in[2];   // [128,128,3,3]
    const float* bias   = (const float*)d_in[3];   // [128]
    const float* x2     = (const float*)d_in[4];   // [8,128,56,56]
    float* out = (float*)d_out;

    float* wP = (float*)d_ws;                      // 9*64*128*2*4 = 576 KB scratch

    const int wtotal = KTAPS * CIN * COUT;         // 147456
    wtrans_kernel<<<(wtotal + 255) / 256, 256, 0, stream>>>(weight, wP);

    const int tiles = (N_ * HWO) / TILE_M;         // 1568
    deform_conv_add_relu_kernel<<<tiles, 256, 0, stream>>>(x, offset, wP, bias, x2, out);
}